// CTCLayer_2551210574586
// MI455X (gfx1250) — compile-verified
//
#include <hip/hip_runtime.h>
#include <stdint.h>

// Problem constants (match reference)
#define Bsz   512
#define Tlen  512
#define Cn    128
#define Llab  32
#define BLANK 127
#define EPSF  1e-7f
#define NEGF  -1e30f

// Streaming chunk: 32 timesteps x 128 floats = 16KB per buffer, double buffered.
#define CH    32
#define NCH   (Tlen / CH)

// 2-way logaddexp (used only at the tail / state 64 path).
__device__ __forceinline__ float lae2(float a, float b) {
    float m = fmaxf(a, b);
    float d = fminf(a, b) - m;           // <= 0
    return m + __logf(1.0f + __expf(d));
}

// Fused 3-way logaddexp: one v_max3_f32, three independent v_exp_f32,
// one v_log_f32 -> much shorter serial chain than nested 2-way.
__device__ __forceinline__ float lae3(float a, float b, float c) {
    float m = fmaxf(fmaxf(a, b), c);     // v_max3_f32
    float s = __expf(a - m) + __expf(b - m) + __expf(c - m);
    return m + __logf(s);
}

__global__ __launch_bounds__(32) void ctc_fwd_kernel(const int* __restrict__ yt,
                                                     const float* __restrict__ yp,
                                                     float* __restrict__ out) {
    __shared__ float pbuf[2][CH * Cn];   // 2 x 16KB

    const int b    = blockIdx.x;
    const int lane = threadIdx.x;

    // Scalar (per-block) global base for GVS async loads.
    const uint64_t gbase = (uint64_t)(uintptr_t)(yp + (size_t)b * Tlen * Cn);
    const uint32_t lb0   = (uint32_t)(uintptr_t)(&pbuf[0][0]);
    const uint32_t lb1   = (uint32_t)(uintptr_t)(&pbuf[1][0]);

    // Labels: lane j owns label j.
    const int lab = yt[b * Llab + lane];

    // State mapping: A0 <-> s=lane, A1 <-> s=lane+32, A2 <-> s=64 (replicated).
    // Odd states s=2j+1 carry label j; even states carry blank.
    const int j0 = (lane - 1) >> 1;        // label idx for s=lane     (odd lanes)
    const int j1 = (lane + 31) >> 1;       // label idx for s=lane+32  (odd lanes)

    const int labj0  = __shfl(lab, j0 & 31, 32);
    const int labj0m = __shfl(lab, (j0 - 1) & 31, 32);
    const int labj1  = __shfl(lab, j1 & 31, 32);
    const int labj1m = __shfl(lab, (j1 - 1) & 31, 32);
    const bool skip0 = (lane & 1) && (j0 >= 1) && (labj0 != BLANK) && (labj0 != labj0m);
    const bool skip1 = (lane & 1) && (labj1 != BLANK) && (labj1 != labj1m);

    // Async copy of one chunk (32 rows x 512B) into LDS buffer at lbase.
    // Each instruction: 32 lanes x 16B = one 512B row. 32 in flight (< 63 cap).
    auto prefetch = [&](uint32_t lbase, int chunk) {
        uint32_t loff = lbase + (uint32_t)lane * 16u;
        uint32_t voff = (uint32_t)chunk * (uint32_t)(CH * Cn * 4) + (uint32_t)lane * 16u;
#pragma unroll
        for (int r = 0; r < CH; ++r) {
            asm volatile("global_load_async_to_lds_b128 %0, %1, %2"
                         :: "v"(loff), "v"(voff), "s"(gbase)
                         : "memory");
            loff += (uint32_t)(Cn * 4);
            voff += (uint32_t)(Cn * 4);
        }
    };

    float A0 = NEGF, A1 = NEGF, A2 = NEGF;

    // Gather the 33 needed log-probs for one timestep row.
    auto gather_lp = [&](const float* row, float& lp0, float& lp1, float& lpB) {
        float lpl = __logf(row[lab]   + EPSF);  // lane j: log p(label_j)
        lpB       = __logf(row[BLANK] + EPSF);  // uniform across lanes
        float g0  = __shfl(lpl, j0 & 31, 32);
        float g1  = __shfl(lpl, j1 & 31, 32);
        lp0 = (lane & 1) ? g0 : lpB;
        lp1 = (lane & 1) ? g1 : lpB;
    };

    // One DP step (t >= 1).
    auto step = [&](const float* row) {
        float lp0, lp1, lpB;
        gather_lp(row, lp0, lp1, lpB);

        // Read all old-alpha neighbors before overwriting.
        float bc31 = __shfl(A0, 31, 32);        // alpha[31]
        float bc30 = __shfl(A0, 30, 32);        // alpha[30]
        float a63  = __shfl(A1, 31, 32);        // alpha[63]

        float s1_0 = __shfl_up(A0, 1, 32); if (lane < 1) s1_0 = NEGF;
        float s2_0 = __shfl_up(A0, 2, 32); if (lane < 2) s2_0 = NEGF;

        float u1   = __shfl_up(A1, 1, 32);
        float u2   = __shfl_up(A1, 2, 32);
        float s1_1 = (lane == 0) ? bc31 : u1;                        // alpha[l+31]
        float s2_1 = (lane == 0) ? bc30 : ((lane == 1) ? bc31 : u2); // alpha[l+30]

        float x2_0 = skip0 ? s2_0 : NEGF;
        float x2_1 = skip1 ? s2_1 : NEGF;

        float n0 = lae3(A0, s1_0, x2_0) + lp0;
        float n1 = lae3(A1, s1_1, x2_1) + lp1;
        float n2 = lae2(A2, a63) + lpB;         // s=64: blank, no skip
        A0 = n0; A1 = n1; A2 = n2;
    };

    // ---- pipeline ----
    prefetch(lb0, 0);

    // Chunk 0: peel t=0 (init) out of the hot loop.
    asm volatile("s_wait_asynccnt 0" ::: "memory");
    prefetch(lb1, 1);
    {
        float lp0, lp1, lpB;
        gather_lp(&pbuf[0][0], lp0, lp1, lpB);
        A0 = (lane < 2) ? lp0 : NEGF;           // alpha0: s=0 blank, s=1 label0
        A1 = NEGF;
        A2 = NEGF;
    }
    for (int i = 1; i < CH; ++i) step(&pbuf[0][i * Cn]);

    // Chunks 1 .. NCH-1.
    for (int c = 1; c < NCH; ++c) {
        asm volatile("s_wait_asynccnt 0" ::: "memory");
        if (c + 1 < NCH) prefetch((c & 1) ? lb0 : lb1, c + 1);
        const float* rowbase = (c & 1) ? &pbuf[1][0] : &pbuf[0][0];
        for (int i = 0; i < CH; ++i) step(rowbase + i * Cn);
    }

    float a63  = __shfl(A1, 31, 32);
    float loss = -lae2(A2, a63);
    if (lane == 0) out[b] = loss;
}

extern "C" void kernel_launch(void* const* d_in, const int* in_sizes, int n_in,
                              void* d_out, int out_size, void* d_ws, size_t ws_size,
                              hipStream_t stream) {
    (void)in_sizes; (void)n_in; (void)out_size; (void)d_ws; (void)ws_size;
    const int*   yt  = (const int*)d_in[0];    // y_true [B, L] int32
    const float* yp  = (const float*)d_in[1];  // y_pred [B, T, C] float32
    float*       out = (float*)d_out;          // [B, 1] float32
    hipLaunchKernelGGL(ctc_fwd_kernel, dim3(Bsz), dim3(32), 0, stream, yt, yp, out);
}